// RLSE_37022618092151
// MI455X (gfx1250) — compile-verified
//
#include <hip/hip_runtime.h>

// CDNA5 / gfx1250: wave32, WMMA f32 16x16x4, 320KB LDS per WGP.
typedef __attribute__((ext_vector_type(2))) float v2f;
typedef __attribute__((ext_vector_type(8))) float v8f;

#define NVAR 256
#define BLK  16
#define NT   256   // 8 wave32 waves on one WGP

__device__ __forceinline__ v8f wmma_f32(v2f a, v2f b, v8f c) {
  // D = A(16x4) * B(4x16) + C(16x16), all f32
  return __builtin_amdgcn_wmma_f32_16x16x4_f32(false, a, false, b, (short)0, c,
                                               false, false);
}

__global__ __launch_bounds__(NT, 1)
void rlse_block_wmma(const float* __restrict__ x, const float* __restrict__ y,
                     const float* __restrict__ s0, const float* __restrict__ t0,
                     float* __restrict__ out, int n_rows) {
  extern __shared__ float smem[];
  float* Sx = smem;            // 65536 : S (256x256, row-major) — whole matrix LDS-resident
  float* Xs = Sx + 65536;      // 4096  : X block (16x256, row-major)
  float* Gs = Xs + 4096;       // 4096  : G = S X^T (256x16)
  float* Hs = Gs + 4096;       // 4096  : H = -g16 * G C (256x16)
  float* Ms = Hs + 4096;       // 256   : M = X S X^T (16x16)
  float* Cs = Ms + 256;        // 256   : inner-space coefficient C (16x16)
  float* th = Cs + 256;        // 256   : theta
  float* us = th + 256;        // 16
  float* dv = us + 16;         // 16    : d (theta = theta0 + G d)
  float* zv = dv + 16;         // 16    : z = X theta0
  float* yv = zv + 16;         // 16
  float* sc = yv + 16;         // 8     : broadcast scalars

  const int tid  = threadIdx.x;
  const int w    = tid >> 5;        // wave id 0..7
  const int lane = tid & 31;
  const int half = lane >> 4;       // 0: lanes 0-15, 1: lanes 16-31
  const int lm   = lane & 15;

  const float gamma = 0.99f;
  float g16 = 1.0f;
#pragma unroll
  for (int i = 0; i < 16; ++i) g16 *= gamma;

  // ---- init state from inputs
  for (int idx = tid; idx < NVAR * NVAR; idx += NT) Sx[idx] = s0[idx];
  th[tid] = t0[tid];
  __syncthreads();

  const int nb = n_rows / BLK;
  for (int b = 0; b < nb; ++b) {
    // ================= region A: load block, reset small state ==============
    const float* xb = x + (size_t)b * BLK * NVAR;
    {
      const float4* src = (const float4*)xb;
      float4*       dst = (float4*)Xs;
      for (int idx = tid; idx < BLK * NVAR / 4; idx += NT) dst[idx] = src[idx];
    }
    if (tid < BLK) yv[tid] = y[b * BLK + tid];
    Cs[tid] = 0.0f;
    if (tid < BLK) dv[tid] = 0.0f;
    if (b + 1 < nb) {
      // one 64B chunk per lane -> covers the whole next 16KB block
      __builtin_prefetch(x + (size_t)(b + 1) * BLK * NVAR + tid * 16, 0, 1);
    }
    __syncthreads();

    // ================= region B: G = S * X^T (256x16), 2 tiles/wave =========
    for (int i = 0; i < 2; ++i) {
      const int m0 = (2 * w + i) * 16;
      v8f acc = {};
      for (int k0 = 0; k0 < NVAR; k0 += 4) {
        v2f a  = *(const v2f*)&Sx[(m0 + lm) * NVAR + k0 + 2 * half];
        v2f bb = *(const v2f*)&Xs[lm * NVAR + k0 + 2 * half]; // B[K][n]=X[n][K]
        acc = wmma_f32(a, bb, acc);
      }
#pragma unroll
      for (int v = 0; v < 8; ++v) Gs[(m0 + v + 8 * half) * BLK + lm] = acc[v];
    }
    // z = X * theta (16 dots of 256)
    if (tid < BLK) {
      float a = 0.0f;
      for (int c = 0; c < NVAR; ++c) a += Xs[tid * NVAR + c] * th[c];
      zv[tid] = a;
    }
    __syncthreads();

    // ================= region C: M = X * G (16x16), wave 0 ==================
    if (w == 0) {
      v8f acc = {};
      for (int k0 = 0; k0 < NVAR; k0 += 4) {
        v2f a = *(const v2f*)&Xs[lm * NVAR + k0 + 2 * half];
        v2f bb;
        bb[0] = Gs[(k0 + 2 * half) * BLK + lm];
        bb[1] = Gs[(k0 + 2 * half + 1) * BLK + lm];
        acc = wmma_f32(a, bb, acc);
      }
#pragma unroll
      for (int v = 0; v < 8; ++v) Ms[(v + 8 * half) * BLK + lm] = acc[v];
    }
    __syncthreads();

    // ======= region D: 16 exact sequential steps in 16-dim space (wave 0) ===
    // S_j = g^j (S - G C_j G^T); u_j=(I-C_j M)e_j; denom=g+g^j M[:,j]^T u;
    // C += g^j/denom * u u^T ; d += g^j*err/denom * u ; err=y_j - z_j - d^T M[:,j]
    if (w == 0) {
      float gj = 1.0f;
      for (int j = 0; j < BLK; ++j) {
        if (lane < BLK) {
          float cu = 0.0f;
#pragma unroll
          for (int t = 0; t < BLK; ++t) cu += Cs[lane * BLK + t] * Ms[t * BLK + j];
          us[lane] = ((lane == j) ? 1.0f : 0.0f) - cu;
        }
        __builtin_amdgcn_wave_barrier();  // DS ops are in-order within a wave
        if (lane == 0) {
          float vtu = 0.0f, ddv = 0.0f;
#pragma unroll
          for (int t = 0; t < BLK; ++t) {
            float v_ = Ms[t * BLK + j];
            vtu += v_ * us[t];
            ddv += dv[t] * v_;
          }
          float denom = gamma + gj * vtu;
          float err   = yv[j] - zv[j] - ddv;
          float scl   = gj / denom;
          sc[0] = scl;
          sc[1] = scl * err;
        }
        __builtin_amdgcn_wave_barrier();
        if (lane < BLK) {
          float s0_ = sc[0], s1_ = sc[1];
          float ul  = us[lane];
#pragma unroll
          for (int t = 0; t < BLK; ++t) Cs[lane * BLK + t] += s0_ * ul * us[t];
          dv[lane] += s1_ * ul;
        }
        __builtin_amdgcn_wave_barrier();
        gj *= gamma;
      }
    }
    __syncthreads();

    // ====== region E: H = -g16*G*C ; S = g16*S + H*G^T ; theta += G*d =======
    for (int i = 0; i < 2; ++i) {
      const int m0 = (2 * w + i) * 16;
      v8f acc = {};
#pragma unroll
      for (int k0 = 0; k0 < BLK; k0 += 4) {
        v2f a = *(const v2f*)&Gs[(m0 + lm) * BLK + k0 + 2 * half];
        v2f bb;
        bb[0] = Cs[(k0 + 2 * half) * BLK + lm];
        bb[1] = Cs[(k0 + 2 * half + 1) * BLK + lm];
        acc = wmma_f32(a, bb, acc);
      }
#pragma unroll
      for (int v = 0; v < 8; ++v)
        Hs[(m0 + v + 8 * half) * BLK + lm] = -g16 * acc[v];
    }
    // rank-16 update of S: each wave owns its own m-row slab (reads its own Hs
    // rows written above -> same-wave DS ordering, no barrier needed here)
    for (int i = 0; i < 2; ++i) {
      const int m0 = (2 * w + i) * 16;
      for (int ct = 0; ct < 16; ++ct) {
        const int c0 = ct * 16;
        v8f acc;
#pragma unroll
        for (int v = 0; v < 8; ++v)
          acc[v] = g16 * Sx[(m0 + v + 8 * half) * NVAR + c0 + lm];
#pragma unroll
        for (int k0 = 0; k0 < BLK; k0 += 4) {
          v2f a  = *(const v2f*)&Hs[(m0 + lm) * BLK + k0 + 2 * half];
          v2f bb = *(const v2f*)&Gs[(c0 + lm) * BLK + k0 + 2 * half]; // G^T chunk
          acc = wmma_f32(a, bb, acc);
        }
#pragma unroll
        for (int v = 0; v < 8; ++v)
          Sx[(m0 + v + 8 * half) * NVAR + c0 + lm] = acc[v];
      }
    }
    // theta += G d (row per thread)
    {
      float a = th[tid];
#pragma unroll
      for (int j = 0; j < BLK; ++j) a += Gs[tid * BLK + j] * dv[j];
      th[tid] = a;
    }
    __syncthreads();
  }

  out[tid] = th[tid];
}

extern "C" void kernel_launch(void* const* d_in, const int* in_sizes, int n_in,
                              void* d_out, int out_size, void* d_ws, size_t ws_size,
                              hipStream_t stream) {
  const float* x  = (const float*)d_in[0];
  const float* y  = (const float*)d_in[1];
  const float* s0 = (const float*)d_in[2];
  const float* t0 = (const float*)d_in[3];
  float* out = (float*)d_out;
  const int n_rows = in_sizes[1];  // 131072

  const size_t smem_floats =
      65536 + 3 * 4096 + 3 * 256 + 4 * 16 + 8;  // ~307 KB < 320 KB WGP LDS
  const size_t smem = smem_floats * sizeof(float);

  (void)hipFuncSetAttribute((const void*)rlse_block_wmma,
                            hipFuncAttributeMaxDynamicSharedMemorySize,
                            (int)smem);

  rlse_block_wmma<<<1, NT, smem, stream>>>(x, y, s0, t0, out, n_rows);
}